// OverlapCalculator_8358006358547
// MI455X (gfx1250) — compile-verified
//
#include <hip/hip_runtime.h>
#include <hip/hip_bf16.h>

#define BLOCK 256
#define ITEMS 2
#define MAXM  2048   // capacity for GT boxes in LDS (32 KB of float4)

#if defined(__gfx1250__) && __has_builtin(__builtin_amdgcn_global_load_async_to_lds_b128)
#define USE_ASYNC_LDS 1
// Match the builtin's parameter types exactly (from hipcc diagnostic):
//   param 0: int __attribute__((vector_size(16))) __device__*  (global src)
//   param 1: LDS destination (addrspace(3))
typedef int v4i __attribute__((vector_size(16)));
typedef __attribute__((address_space(1))) v4i as1_v4i;
typedef __attribute__((address_space(3))) v4i as3_v4i;
#else
#define USE_ASYNC_LDS 0
#endif

__global__ __launch_bounds__(BLOCK) void iou_max_kernel(
    const float* __restrict__ pred,   // [N,4] xyxy
    const float* __restrict__ gt,     // [M,4] xyxy
    float* __restrict__ out,          // pred(4N) | gt(4M) | max(N) | argmax(N)
    int N, int M)
{
    __shared__ float4 sgt[MAXM];

    // ---- Stage GT boxes into LDS (CDNA5 async global->LDS path) ----
#if USE_ASYNC_LDS
    for (int j = threadIdx.x; j < M; j += BLOCK) {
        __builtin_amdgcn_global_load_async_to_lds_b128(
            (as1_v4i*)(gt + 4 * j), (as3_v4i*)(&sgt[j]), 0, 0);
    }
    asm volatile("s_wait_asynccnt 0" ::: "memory");
#else
    for (int j = threadIdx.x; j < M; j += BLOCK) {
        sgt[j] = ((const float4*)gt)[j];
    }
#endif
    __syncthreads();

    const int t  = blockIdx.x * BLOCK + threadIdx.x;
    const int i0 = t * ITEMS;
    const int i1 = i0 + 1;
    const bool v0 = (i0 < N);
    const bool v1 = (i1 < N);

    // GT passthrough copy (first M threads)
    if (t < M) {
        ((float4*)(out + 4 * N))[t] = sgt[t];
    }

    float4 p0 = v0 ? ((const float4*)pred)[i0] : make_float4(0.f, 0.f, 0.f, 0.f);
    float4 p1 = v1 ? ((const float4*)pred)[i1] : make_float4(0.f, 0.f, 0.f, 0.f);
    const float ap0 = (p0.z - p0.x) * (p0.w - p0.y);
    const float ap1 = (p1.z - p1.x) * (p1.w - p1.y);

    // Running best as (inter, union) pair; compare via cross-multiplication
    // (inter >= 0, union > 0) -> no division in the hot loop, and strict '>'
    // preserves first-occurrence argmax like jnp.argmax.
    float bi0 = 0.f, bu0 = 1.f;
    float bi1 = 0.f, bu1 = 1.f;
    int   bj0 = 0,   bj1 = 0;

#pragma unroll 4
    for (int j = 0; j < M; ++j) {
        const float4 g  = sgt[j];                       // ds_load_b128 broadcast
        const float  ag = (g.z - g.x) * (g.w - g.y);    // shared by both preds

        // pred 0
        {
            const float ltx = fmaxf(p0.x, g.x);
            const float lty = fmaxf(p0.y, g.y);
            const float rbx = fminf(p0.z, g.z);
            const float rby = fminf(p0.w, g.w);
            const float w   = fmaxf(rbx - ltx, 0.f);
            const float h   = fmaxf(rby - lty, 0.f);
            const float inter = w * h;
            const float uni   = ap0 + ag - inter;
            const bool better = (inter * bu0) > (bi0 * uni);
            bi0 = better ? inter : bi0;
            bu0 = better ? uni   : bu0;
            bj0 = better ? j     : bj0;
        }
        // pred 1
        {
            const float ltx = fmaxf(p1.x, g.x);
            const float lty = fmaxf(p1.y, g.y);
            const float rbx = fminf(p1.z, g.z);
            const float rby = fminf(p1.w, g.w);
            const float w   = fmaxf(rbx - ltx, 0.f);
            const float h   = fmaxf(rby - lty, 0.f);
            const float inter = w * h;
            const float uni   = ap1 + ag - inter;
            const bool better = (inter * bu1) > (bi1 * uni);
            bi1 = better ? inter : bi1;
            bu1 = better ? uni   : bu1;
            bj1 = better ? j     : bj1;
        }
    }

    float* outMax = out + 4 * N + 4 * M;
    float* outArg = outMax + N;

    if (v0) {
        ((float4*)out)[i0] = p0;          // pred passthrough
        outMax[i0] = bi0 / bu0;           // single division per row
        outArg[i0] = (float)bj0;
    }
    if (v1) {
        ((float4*)out)[i1] = p1;
        outMax[i1] = bi1 / bu1;
        outArg[i1] = (float)bj1;
    }
}

extern "C" void kernel_launch(void* const* d_in, const int* in_sizes, int n_in,
                              void* d_out, int out_size, void* d_ws, size_t ws_size,
                              hipStream_t stream) {
    const float* pred = (const float*)d_in[0];
    const float* gt   = (const float*)d_in[1];
    float* out        = (float*)d_out;

    const int N = in_sizes[0] / 4;   // 50000
    const int M = in_sizes[1] / 4;   // 2000 (must be <= MAXM)

    const int threads_needed = (N + ITEMS - 1) / ITEMS;
    const int blocks = (threads_needed + BLOCK - 1) / BLOCK;

    iou_max_kernel<<<blocks, BLOCK, 0, stream>>>(pred, gt, out, N, M);
}